// Attention_10299331576042
// MI455X (gfx1250) — compile-verified
//
#include <hip/hip_runtime.h>
#include <math.h>

#define H 1024
#define S 32768

typedef float v2f __attribute__((ext_vector_type(2)));
typedef float v8f __attribute__((ext_vector_type(8)));

// ---------------------------------------------------------------------------
// Phase 1: v[h] = sum_d W[d][h] * hidden[d]   (v = W^T @ hidden)
// 16 blocks x 256 threads. Block b owns columns [64b, 64b+64).
// Thread t: hloc = t&63 (column), dpart = t>>6 (one of 4 slices of d).
// W loads are 64 consecutive floats per dpart-group -> 256B coalesced.
// ---------------------------------------------------------------------------
__global__ void gemv_wt_h(const float* __restrict__ W,
                          const float* __restrict__ hidden,
                          float* __restrict__ v) {
  __shared__ __align__(16) float sh[H];
  __shared__ __align__(16) float sred[256];
  const int t = threadIdx.x;
  for (int i = t; i < H; i += 256) sh[i] = hidden[i];
  __syncthreads();

  const int hloc  = t & 63;
  const int dpart = t >> 6;
  const int hcol  = blockIdx.x * 64 + hloc;
  const int dbase = dpart * 256;

  float acc = 0.0f;
#pragma unroll 4
  for (int d = 0; d < 256; ++d) {
    acc = fmaf(W[(size_t)(dbase + d) * H + hcol], sh[dbase + d], acc);
  }
  sred[t] = acc;
  __syncthreads();
  if (t < 64) {
    v[blockIdx.x * 64 + t] = sred[t] + sred[t + 64] + sred[t + 128] + sred[t + 192];
  }
}

// ---------------------------------------------------------------------------
// Phase 2: scores[s] = enc[s,:] . v   via V_WMMA_F32_16X16X4_F32.
// 256 blocks x 256 threads (8 waves). Each wave owns one 16-row tile of enc.
// B's 16 columns are all the same v-chunk, so every D column holds the same
// 16 scores; waste is pure compute, and the kernel stays on the 23.3 TB/s
// bandwidth roofline (128 MB of enc is the only real traffic).
//
// ISA layouts (CDNA5 ISA):
//   A 16x4 f32 : lane L(0-15) -> M=L, VGPR{0,1}=K{0,1}; lane L+16 -> K{2,3}
//   B 4x16 f32 : VGPR{0,1} = rows K{0,1} (lanes 0-15), K{2,3} (lanes 16-31)
//   D 16x16 f32: lane n(0-15) VGPR r = D[r][n]; lane n+16 VGPR r = D[r+8][n]
// ---------------------------------------------------------------------------
__global__ void scores_wmma(const float* __restrict__ enc,
                            const float* __restrict__ v,
                            float* __restrict__ scores) {
  __shared__ __align__(16) float sv[H];
  const int t = threadIdx.x;
  for (int i = t; i < H; i += 256) sv[i] = v[i];
  __syncthreads();

  const int wave    = t >> 5;
  const int lane    = t & 31;
  const int hi      = lane >> 4;   // 0: lanes 0-15, 1: lanes 16-31
  const int m15     = lane & 15;
  const int rowbase = (blockIdx.x * 8 + wave) << 4;   // *16 rows

  const float* rowptr = enc + (size_t)(rowbase + m15) * H + 2 * hi;

  v8f acc = {0.f, 0.f, 0.f, 0.f, 0.f, 0.f, 0.f, 0.f};
#pragma unroll 8
  for (int k = 0; k < H; k += 4) {
    v2f a = *reinterpret_cast<const v2f*>(rowptr + k);          // enc[m][k+2hi .. +1]
    v2f b = *reinterpret_cast<const v2f*>(&sv[k + 2 * hi]);     // v   [k+2hi .. +1]
    acc = __builtin_amdgcn_wmma_f32_16x16x4_f32(
        /*neg_a=*/false, a, /*neg_b=*/false, b,
        /*c_mod=*/(short)0, acc, /*reuse_a=*/false, /*reuse_b=*/false);
  }

  // All D columns identical: lane 0 holds rows rowbase..+7 in acc[0..7],
  // lane 16 holds rows rowbase+8..+15.
  if (m15 == 0) {
    float* out = scores + rowbase + hi * 8;
#pragma unroll
    for (int r = 0; r < 8; ++r) out[r] = acc[r];
  }
}

// ---------------------------------------------------------------------------
// Phase 3a: global max and 1/sum(exp) over the 32768 scores (single block).
// ---------------------------------------------------------------------------
__global__ void softmax_stats(const float* __restrict__ scores,
                              float* __restrict__ stats) {
  __shared__ __align__(16) float red[1024];
  const int t = threadIdx.x;

  float m = -INFINITY;
  for (int i = t; i < S; i += 1024) m = fmaxf(m, scores[i]);
  red[t] = m;
  __syncthreads();
  for (int s = 512; s > 0; s >>= 1) {
    if (t < s) red[t] = fmaxf(red[t], red[t + s]);
    __syncthreads();
  }
  const float gmax = red[0];
  __syncthreads();

  float sum = 0.0f;
  for (int i = t; i < S; i += 1024) sum += expf(scores[i] - gmax);
  red[t] = sum;
  __syncthreads();
  for (int s = 512; s > 0; s >>= 1) {
    if (t < s) red[t] += red[t + s];
    __syncthreads();
  }
  if (t == 0) {
    stats[0] = gmax;
    stats[1] = 1.0f / red[0];
  }
}

// ---------------------------------------------------------------------------
// Phase 3b: out[i] = exp(scores[i]-max) * invsum
// ---------------------------------------------------------------------------
__global__ void softmax_norm(const float* __restrict__ scores,
                             const float* __restrict__ stats,
                             float* __restrict__ out) {
  const int i = blockIdx.x * 256 + threadIdx.x;
  const float gmax = stats[0];
  const float inv  = stats[1];
  out[i] = expf(scores[i] - gmax) * inv;
}

// ---------------------------------------------------------------------------
// Launch. Inputs (setup_inputs order): hidden [H], encoder_outputs [S*H],
// W [H*H], b [H]. Output: [1,1,S] f32.
// Note: the bias contributes b.hidden (a scalar) to every score; softmax is
// shift-invariant so it cancels exactly and is skipped.
// ---------------------------------------------------------------------------
extern "C" void kernel_launch(void* const* d_in, const int* in_sizes, int n_in,
                              void* d_out, int out_size, void* d_ws, size_t ws_size,
                              hipStream_t stream) {
  const float* hidden = (const float*)d_in[0];
  const float* enc    = (const float*)d_in[1];
  const float* W      = (const float*)d_in[2];
  (void)d_in; (void)in_sizes; (void)n_in; (void)out_size; (void)ws_size;

  float* ws     = (float*)d_ws;
  float* v      = ws;            // H floats
  float* scores = ws + H;        // S floats
  float* stats  = ws + H + S;    // 2 floats

  gemv_wt_h   <<<H / 64,        256,  0, stream>>>(W, hidden, v);
  scores_wmma <<<S / (16 * 8),  256,  0, stream>>>(enc, v, scores);
  softmax_stats<<<1,            1024, 0, stream>>>(scores, stats);
  softmax_norm<<<S / 256,       256,  0, stream>>>(scores, stats, (float*)d_out);
}